// SpatialInputLayer_59433757442309
// MI455X (gfx1250) — compile-verified
//
#include <hip/hip_runtime.h>
#include <hip/hip_bf16.h>
#include <math.h>
#include <stdint.h>

// ---------------------------------------------------------------------------
// SpatialInputLayer for MI455X (gfx1250, wave32).
// Canvas build (memory-bound scatters) + 13x13 Gaussian blur expressed as
// Out = G * X * G  (G = 64x64 symmetric band matrix) using f32 WMMA
// (V_WMMA_F32_16X16X4_F32). The image is DMA'd into LDS with
// GLOBAL_LOAD_ASYNC_TO_LDS_B128 (ASYNCcnt), overlapped with building G,
// and the blur runs in place in d_out (no workspace needed).
// Fragment symmetry tricks: G symmetric -> B-fragments are contiguous b64
// LDS loads; T stored transposed -> stage-2 B-fragments contiguous too and
// stage-1 accumulator stores become contiguous b128s.
// ---------------------------------------------------------------------------

#define NCH    16
#define NTOW   30
#define PITCH  68          // LDS row pitch (floats): mult-of-4 -> 8B/16B aligned
                           // pairs, bank stride 4/row -> conflict-free halves
#define SCALE_F 234.375f   // 15000 / 64

typedef __attribute__((ext_vector_type(2))) float v2f;
typedef __attribute__((ext_vector_type(8))) float v8f;

__constant__ float c_towx[NTOW] = {
    8955.f, 9767.f, 11134.f, 11593.f, 13052.f, 12611.f, 5846.f, 5048.f,
    3651.f, 3210.f, 2177.f, 1748.f, 4318.f, 981.f, 7943.f, 1512.f,
    1169.f, 1172.f, 10481.f, 11275.f, 11275.f, 13866.f, 10504.f, 13327.f,
    13624.f, 6919.f, 13599.f, 4281.f, 3468.f, 13594.f};
__constant__ float c_towy[NTOW] = {
    8510.f, 10113.f, 11207.f, 11669.f, 12612.f, 13084.f, 6396.f, 4812.f,
    3696.f, 3217.f, 1807.f, 2270.f, 13875.f, 10441.f, 13411.f, 6699.f,
    4287.f, 3583.f, 13650.f, 13657.f, 13663.f, 4505.f, 1029.f, 8226.f,
    10572.f, 1483.f, 11319.f, 1253.f, 1230.f, 11319.f};

__device__ __forceinline__ int clampi(int v, int lo, int hi) {
    return v < lo ? lo : (v > hi ? hi : v);
}

// ---------------------------------------------------------------------------
// Kernel 1: per-batch base channels. One block per batch element.
// Writes the entire 16x64x64 slice of d_out (ch0..3 computed, ch4..15 zero).
// ---------------------------------------------------------------------------
__global__ __launch_bounds__(256) void k_base(const float* __restrict__ dead_mask,
                                              const float* __restrict__ obj_status,
                                              float* __restrict__ out) {
    const int b   = blockIdx.x;
    const int tid = threadIdx.x;

    __shared__ int   stx[NTOW];
    __shared__ int   sty[NTOW];
    __shared__ float salive[NTOW];
    __shared__ float sobj[2];

    if (tid < NTOW) {
        // match numpy: float32 divide, truncate, clamp to [0,63]
        stx[tid]    = clampi((int)(c_towx[tid] / SCALE_F), 0, 63);
        sty[tid]    = clampi((int)(c_towy[tid] / SCALE_F), 0, 63);
        salive[tid] = (dead_mask[b * NTOW + tid] > 0.5f) ? 0.0f : 1.0f;
    }
    if (tid < 2) sobj[tid] = (obj_status[2 * b + tid] > 0.5f) ? 1.0f : 0.0f;
    __syncthreads();

    // DRAG=(x=42,y=21)  BARON=(x=21,y=42)  (int(10000/234.375), int(5000/234.375))
    const int DX = 42, DY = 21, BX = 21, BY = 42;

    float* base = out + (size_t)b * NCH * 4096;
    for (int cell = tid; cell < 4096; cell += 256) {
        const int x = cell & 63;
        const int y = cell >> 6;
        float c0 = 0.f, c1 = 0.f, c2 = 0.f;
        #pragma unroll
        for (int t = 0; t < NTOW; ++t) {
            const int dx = x - stx[t];
            const int dy = y - sty[t];
            const float al = salive[t];
            if (dx * dx + dy * dy <= 16 && al > 0.f) c2 = 1.f;        // R=4 in grid units
            if (dx == 0 && dy == 0) {
                c0 = fmaxf(c0, al);
                c1 = fmaxf(c1, 1.f - al);
            }
        }
        float c3 = 0.f;
        if (y == DY && x == DX) c3 = sobj[0];
        if (y == BY && x == BX) c3 = sobj[1];

        base[0 * 4096 + cell] = c0;
        base[1 * 4096 + cell] = c1;
        base[2 * 4096 + cell] = c2;
        base[3 * 4096 + cell] = c3;
        #pragma unroll
        for (int c = 4; c < NCH; ++c) base[c * 4096 + cell] = 0.f;
    }
}

// ---------------------------------------------------------------------------
// Kernel 2: scatter-add of point events (kills: chan_base=4, trails: 6).
// ---------------------------------------------------------------------------
__global__ __launch_bounds__(256) void k_scatter(const float* __restrict__ coords,
                                                 const float* __restrict__ vals,
                                                 const int* __restrict__ bidx,
                                                 const int* __restrict__ chan,
                                                 int chan_base, int n,
                                                 float* __restrict__ out) {
    const int i = blockIdx.x * 256 + threadIdx.x;
    if (i >= n) return;
    const int gx = clampi((int)(coords[2 * i + 0] / SCALE_F), 0, 63);
    const int gy = clampi((int)(coords[2 * i + 1] / SCALE_F), 0, 63);
    const int c  = chan_base + chan[i];
    const size_t off = ((size_t)bidx[i] * NCH + c) * 4096 + (size_t)(gy * 64 + gx);
    atomicAdd(out + off, vals[i]);
}

// ---------------------------------------------------------------------------
// Kernel 3: 13x13 Gaussian blur as  Out = G * X * G  with f32 WMMA.
// One 256-thread block (8 wave32) per (b,c) image; in-place on d_out.
//   A fragment (16x4 f32): lane l, reg v -> A[l&15][kk + v + 2*(l>>4)]
//   B fragment (4x16 f32): lane l, reg v -> B[kk + v + 2*(l>>4)][l&15]
//   D (16x16 f32):         lane l, reg v -> D[16*ti + v + 8*(l>>4)][16*tj + (l&15)]
// G is symmetric and sT holds T transposed, so every fragment pair is a
// contiguous (8B-aligned) LDS load and acc stores are contiguous b128s.
// ---------------------------------------------------------------------------
__global__ __launch_bounds__(256) void k_blur(float* __restrict__ out) {
    __shared__ float sX[64 * PITCH];   // source image (row-major, pitch 68)
    __shared__ float sT[64 * PITCH];   // T^T  (sT[c*PITCH+r] = T[r][c])
    __shared__ float sG[64 * PITCH];   // Gaussian band matrix (symmetric)

    const int tid = threadIdx.x;
    float* img = out + (size_t)blockIdx.x * 4096;

    // ---- async DMA the whole image into LDS (ASYNCcnt), 16B per op -------
    // 4096 floats = 1024 b128 chunks; 256 threads x 4 chunks.
    for (int v4 = tid; v4 < 1024; v4 += 256) {
        const int row = v4 >> 4;            // 16 chunks per 64-float row
        const int col = (v4 & 15) << 2;
        const uint32_t ldsoff =
            (uint32_t)(uintptr_t)&sX[row * PITCH + col];  // low 32b = LDS offset
        const float* gp = img + (row * 64 + col);
        asm volatile("global_load_async_to_lds_b128 %0, %1, off"
                     :: "v"(ldsoff), "v"(gp)
                     : "memory");
    }

    // ---- build G while the DMA is in flight ------------------------------
    // Normalized 1D Gaussian, sigma=1.5, taps=13: w[i]=exp(-d^2/4.5)/sum
    float w[13];
    float s = 0.f;
    #pragma unroll
    for (int i = 0; i < 13; ++i) {
        const float d = (float)(i - 6);
        w[i] = expf(-d * d / 4.5f);
        s += w[i];
    }
    const float invs = 1.f / s;
    for (int idx = tid; idx < 4096; idx += 256) {
        const int r = idx >> 6, c = idx & 63;
        const int d = c - r;
        sG[r * PITCH + c] = (d >= -6 && d <= 6) ? w[d + 6] * invs : 0.f;
    }

    asm volatile("s_wait_asynccnt 0" ::: "memory");
    __syncthreads();

    const int lane  = tid & 31;
    const int wave  = tid >> 5;
    const int m     = lane & 15;       // fragment row/col within tile
    const int khalf = (lane >> 4) * 2; // K sub-offset per half-wave
    const int rhalf = (lane >> 4) * 8; // D row sub-offset per half-wave

    // ---------------- Stage 1: T = X * G  (store T^T) ----------------
    for (int tt = 2 * wave; tt < 2 * wave + 2; ++tt) {
        const int ti = tt >> 2, tj = tt & 3;
        const int ar = ti * 16 + m;    // row of X
        const int bc = tj * 16 + m;    // col of G
        v8f acc = (v8f)0.0f;
        #pragma unroll
        for (int kk = 0; kk < 64; kk += 4) {
            const int k0 = kk + khalf;
            v2f a  = { sX[ar * PITCH + k0], sX[ar * PITCH + k0 + 1] };
            // G symmetric: G[k][bc] == G[bc][k] -> contiguous pair
            v2f bm = { sG[bc * PITCH + k0], sG[bc * PITCH + k0 + 1] };
            acc = __builtin_amdgcn_wmma_f32_16x16x4_f32(
                false, a, false, bm, (short)0, acc, false, false);
        }
        // store transposed: sT[col*PITCH + row] -> 8 contiguous floats
        const int rbase = ti * 16 + rhalf;
        #pragma unroll
        for (int v = 0; v < 8; ++v) sT[bc * PITCH + rbase + v] = acc[v];
    }
    __syncthreads();

    // ---------------- Stage 2: Out = G * T --------------------------
    for (int tt = 2 * wave; tt < 2 * wave + 2; ++tt) {
        const int ti = tt >> 2, tj = tt & 3;
        const int ar = ti * 16 + m;    // row of G
        const int bc = tj * 16 + m;    // col of T
        v8f acc = (v8f)0.0f;
        #pragma unroll
        for (int kk = 0; kk < 64; kk += 4) {
            const int k0 = kk + khalf;
            v2f a  = { sG[ar * PITCH + k0], sG[ar * PITCH + k0 + 1] };
            // T[k][bc] == sT[bc*PITCH + k] -> contiguous pair
            v2f bm = { sT[bc * PITCH + k0], sT[bc * PITCH + k0 + 1] };
            acc = __builtin_amdgcn_wmma_f32_16x16x4_f32(
                false, a, false, bm, (short)0, acc, false, false);
        }
        const int rbase = ti * 16 + rhalf;
        #pragma unroll
        for (int v = 0; v < 8; ++v) img[(rbase + v) * 64 + bc] = acc[v];
    }
}

// ---------------------------------------------------------------------------
// Host launcher. Input order (setup_inputs):
//  0 player_coords f32 (N_TRAIL,2)   1 player_vals f32 (N_TRAIL)
//  2 dead_towers_mask f32 (B,30)     3 kill_coords f32 (N_KILL,2)
//  4 kill_vals f32 (N_KILL)          5 obj_status f32 (B,2)
//  6 player_b_idx i32                7 player_channels i32
//  8 kill_teams i32                  9 kill_b_idx i32
// ---------------------------------------------------------------------------
extern "C" void kernel_launch(void* const* d_in, const int* in_sizes, int n_in,
                              void* d_out, int out_size, void* d_ws, size_t ws_size,
                              hipStream_t stream) {
    const float* player_coords = (const float*)d_in[0];
    const float* player_vals   = (const float*)d_in[1];
    const float* dead_mask     = (const float*)d_in[2];
    const float* kill_coords   = (const float*)d_in[3];
    const float* kill_vals     = (const float*)d_in[4];
    const float* obj_status    = (const float*)d_in[5];
    const int*   player_b_idx  = (const int*)d_in[6];
    const int*   player_chan   = (const int*)d_in[7];
    const int*   kill_teams    = (const int*)d_in[8];
    const int*   kill_b_idx    = (const int*)d_in[9];
    float*       out           = (float*)d_out;

    const int B       = in_sizes[2] / NTOW;
    const int N_TRAIL = in_sizes[1];
    const int N_KILL  = in_sizes[4];

    k_base<<<B, 256, 0, stream>>>(dead_mask, obj_status, out);
    k_scatter<<<(N_KILL + 255) / 256, 256, 0, stream>>>(
        kill_coords, kill_vals, kill_b_idx, kill_teams, 4, N_KILL, out);
    k_scatter<<<(N_TRAIL + 255) / 256, 256, 0, stream>>>(
        player_coords, player_vals, player_b_idx, player_chan, 6, N_TRAIL, out);
    k_blur<<<B * NCH, 256, 0, stream>>>(out);
}